// RNNAssembly_88570815578415
// MI455X (gfx1250) — compile-verified
//
#include <hip/hip_runtime.h>

// ---------------------------------------------------------------------------
// RNN assembly on MI455X (gfx1250, wave32, WMMA). Sizes fixed by reference.
// ---------------------------------------------------------------------------
#define T_STEPS 2048
#define BATCH   16
#define IN_DIM  128
#define OUT_DIM 128
#define KBLK    32      // number of 64x64 blocks
#define BS      64      // block size
#define HDIM    2048    // KBLK*BS
#define DT      0.01f

typedef __attribute__((ext_vector_type(2))) float v2f;
typedef __attribute__((ext_vector_type(8))) float v8f;

static __device__ __forceinline__ v8f wmma_f32(v2f a, v2f b, v8f c) {
  // 8 args: (neg_a, A, neg_b, B, c_mod, C, reuse_a, reuse_b)
  return __builtin_amdgcn_wmma_f32_16x16x4_f32(false, a, false, b, (short)0, c,
                                               false, false);
}

// ---------------------------------------------------------------------------
// fp32 WMMA GEMM with 2x2 register blocking: each wave computes a 32x32 tile
// (4 accumulators), reusing every A and B fragment twice.
// Fragment layouts (CDNA5 ISA 7.12.2):
//   A: a[v] = A[m = lane%16][k0 + 2*(lane/16) + v]
//   B: b[v] = B[k0 + v + 2*(lane/16)][n = lane%16]
//   D: d[v] = C[m = v + 8*(lane/16)][n = lane%16]
// ---------------------------------------------------------------------------
__global__ void __launch_bounds__(256)
gemm_wmma_f32(const float* __restrict__ A, const float* __restrict__ B,
              float* __restrict__ C, int M, int K, int N) {
  const int lane = threadIdx.x & 31;
  const int wave = threadIdx.x >> 5;
  const int tilesN = N >> 5;                       // 32-wide tiles
  const int tile = blockIdx.x * (blockDim.x >> 5) + wave;
  const int tm = (tile / tilesN) * 32;
  const int tn = (tile % tilesN) * 32;
  if (tm >= M) return;
  const int half = lane >> 4;
  const int l16  = lane & 15;

  v8f acc00 = {}, acc01 = {}, acc10 = {}, acc11 = {};
  const float* A0 = A + (size_t)(tm + l16) * K + half * 2;
  const float* A1 = A0 + (size_t)16 * K;
  const float* B0 = B + (size_t)(half * 2) * N + tn + l16;
  const float* B1 = B0 + 16;
  for (int k0 = 0; k0 < K; k0 += 4) {
    v2f a0, a1, b0, b1;
    a0.x = A0[k0];  a0.y = A0[k0 + 1];             // contiguous -> b64
    a1.x = A1[k0];  a1.y = A1[k0 + 1];
    b0.x = B0[(size_t)k0 * N];  b0.y = B0[(size_t)(k0 + 1) * N];
    b1.x = B1[(size_t)k0 * N];  b1.y = B1[(size_t)(k0 + 1) * N];
    acc00 = wmma_f32(a0, b0, acc00);
    acc01 = wmma_f32(a0, b1, acc01);
    acc10 = wmma_f32(a1, b0, acc10);
    acc11 = wmma_f32(a1, b1, acc11);
  }
#pragma unroll
  for (int v = 0; v < 8; ++v) {
    int m = v + 8 * half;
    float* Crow0 = C + (size_t)(tm + m) * N + tn + l16;
    float* Crow1 = C + (size_t)(tm + 16 + m) * N + tn + l16;
    Crow0[0]  = acc00[v];
    Crow0[16] = acc01[v];
    Crow1[0]  = acc10[v];
    Crow1[16] = acc11[v];
  }
}

// ---------------------------------------------------------------------------
// Sparsity detection: one workgroup per 64x64 block of `coupling`.
// ---------------------------------------------------------------------------
__global__ void __launch_bounds__(256)
find_nz_blocks(const float* __restrict__ C, int* __restrict__ nz) {
  __shared__ int flag;
  if (threadIdx.x == 0) flag = 0;
  __syncthreads();
  const int bi = blockIdx.x / KBLK;
  const int bj = blockIdx.x % KBLK;
  const float* base = C + (size_t)(bi * BS) * HDIM + bj * BS;
  int any = 0;
  for (int e = threadIdx.x; e < BS * BS; e += blockDim.x) {
    int r = e >> 6, c = e & 63;
    if (base[(size_t)r * HDIM + c] != 0.0f) any = 1;
  }
  if (any) atomicOr(&flag, 1);
  __syncthreads();
  if (threadIdx.x == 0) nz[blockIdx.x] = flag;
}

__global__ void build_lists(const int* __restrict__ nz, int* __restrict__ cnt,
                            int* __restrict__ lst) {
  int j = threadIdx.x;
  if (j >= KBLK) return;
  int c = 0;
  for (int i = 0; i < KBLK; ++i)
    if (i != j && nz[i * KBLK + j]) lst[j * KBLK + c++] = i;
  cnt[j] = c;
}

// ---------------------------------------------------------------------------
// Pack block-diagonal weights into exact B-fragment order so the scan's hot
// loop does one coalesced b64 load per fragment:
//   pkd[j][kq][n0][lane][v] = blocks[j][kq*4 + v + 2*(lane/16)][n0*16 + lane%16]
// ---------------------------------------------------------------------------
__global__ void __launch_bounds__(256)
pack_diag(const float* __restrict__ blocks, float* __restrict__ pkd) {
  const int j = blockIdx.x;
  const float* W = blocks + (size_t)j * BS * BS;
  float* P = pkd + (size_t)j * BS * BS;
  for (int e = threadIdx.x; e < BS * BS; e += blockDim.x) {
    int v    = e & 1;
    int lane = (e >> 1) & 31;
    int n0   = (e >> 6) & 3;
    int kq   = e >> 8;
    int row  = kq * 4 + v + 2 * (lane >> 4);
    int col  = n0 * 16 + (lane & 15);
    P[e] = W[row * BS + col];
  }
}

// ---------------------------------------------------------------------------
// The sequential scan. ONE workgroup, 1024 threads = 32 waves = 32 blocks.
// Full state [16 x 2048] fp32 resident in LDS (128 KB of the WGP's 320 KB).
// Wave j owns state columns [64j, 64j+64): four 16x16 WMMA accumulators.
// emb holds inp_proj on entry; overwritten in place with embeddings.
// ---------------------------------------------------------------------------
__global__ void __launch_bounds__(1024)
scan_kernel(const float* __restrict__ pkd,
            const float* __restrict__ coupling,
            float* __restrict__ emb,
            const int* __restrict__ cnt, const int* __restrict__ lst) {
  __shared__ float state[BATCH * HDIM];   // 128 KB
  __shared__ int s_cnt[KBLK];
  __shared__ int s_lst[KBLK * KBLK];

  for (int e = threadIdx.x; e < BATCH * HDIM; e += blockDim.x) state[e] = 0.0f;
  for (int e = threadIdx.x; e < KBLK; e += blockDim.x) s_cnt[e] = cnt[e];
  for (int e = threadIdx.x; e < KBLK * KBLK; e += blockDim.x) s_lst[e] = lst[e];
  __syncthreads();

  const int lane = threadIdx.x & 31;
  const int j    = threadIdx.x >> 5;   // wave id == output block id
  const int half = lane >> 4;
  const int l16  = lane & 15;
  const int nnz  = s_cnt[j];

  // Packed diag fragments for this wave: one v2f per (kq, n0) per lane.
  const v2f* Pd = (const v2f*)(pkd + (size_t)j * BS * BS) + lane;

  for (int t = 0; t < T_STEPS; ++t) {
    float* u = emb + (size_t)t * BATCH * HDIM;

    // ---- read phase: init acc with u, accumulate blk + cpl ----------------
    v8f acc0 = {}, acc1 = {}, acc2 = {}, acc3 = {};
#pragma unroll
    for (int v = 0; v < 8; ++v) {
      int m = v + 8 * half;
      size_t ro = (size_t)m * HDIM + j * BS + l16;
      acc0[v] = u[ro + 0];
      acc1[v] = u[ro + 16];
      acc2[v] = u[ro + 32];
      acc3[v] = u[ro + 48];
    }
    // Block-diagonal term: A = relu(state block j), packed weights.
    {
      const float* As = state + (size_t)l16 * HDIM + j * BS + half * 2;
#pragma unroll
      for (int kq = 0; kq < 16; ++kq) {
        v2f a;
        a.x = fmaxf(As[kq * 4 + 0], 0.0f);
        a.y = fmaxf(As[kq * 4 + 1], 0.0f);
        v2f b0 = Pd[(kq * 4 + 0) * 32];
        v2f b1 = Pd[(kq * 4 + 1) * 32];
        v2f b2 = Pd[(kq * 4 + 2) * 32];
        v2f b3 = Pd[(kq * 4 + 3) * 32];
        acc0 = wmma_f32(a, b0, acc0);
        acc1 = wmma_f32(a, b1, acc1);
        acc2 = wmma_f32(a, b2, acc2);
        acc3 = wmma_f32(a, b3, acc3);
      }
    }
    // Sparse coupling: only nonzero 64x64 blocks in column j (~1.6 avg).
    for (int c = 0; c < nnz; ++c) {
      const int i = s_lst[j * KBLK + c];
      const float* As = state + (size_t)l16 * HDIM + i * BS + half * 2;
      const float* Wc = coupling + (size_t)(i * BS + half * 2) * HDIM + j * BS + l16;
      for (int k0 = 0; k0 < BS; k0 += 4) {
        v2f a;
        a.x = As[k0 + 0];
        a.y = As[k0 + 1];
        const float* wb = Wc + (size_t)k0 * HDIM;
        v2f b0, b1, b2, b3;
        b0.x = wb[0];  b0.y = wb[HDIM];
        b1.x = wb[16]; b1.y = wb[HDIM + 16];
        b2.x = wb[32]; b2.y = wb[HDIM + 32];
        b3.x = wb[48]; b3.y = wb[HDIM + 48];
        acc0 = wmma_f32(a, b0, acc0);
        acc1 = wmma_f32(a, b1, acc1);
        acc2 = wmma_f32(a, b2, acc2);
        acc3 = wmma_f32(a, b3, acc3);
      }
    }
    // Euler update (reads old state -> still read phase).
    float n0[8], n1[8], n2[8], n3[8];
#pragma unroll
    for (int v = 0; v < 8; ++v) {
      int m = v + 8 * half;
      int off = m * HDIM + j * BS + l16;
      n0[v] = (1.0f - DT) * state[off + 0]  + DT * acc0[v];
      n1[v] = (1.0f - DT) * state[off + 16] + DT * acc1[v];
      n2[v] = (1.0f - DT) * state[off + 32] + DT * acc2[v];
      n3[v] = (1.0f - DT) * state[off + 48] + DT * acc3[v];
    }
    __syncthreads();   // all waves done reading old state
    // ---- write phase: new state to LDS + embedding to global --------------
#pragma unroll
    for (int v = 0; v < 8; ++v) {
      int m = v + 8 * half;
      int off = m * HDIM + j * BS + l16;
      state[off + 0]  = n0[v];
      state[off + 16] = n1[v];
      state[off + 32] = n2[v];
      state[off + 48] = n3[v];
      size_t go = (size_t)m * HDIM + j * BS + l16;
      u[go + 0]  = n0[v];
      u[go + 16] = n1[v];
      u[go + 32] = n2[v];
      u[go + 48] = n3[v];
    }
    __syncthreads();   // writes visible before next read phase
  }
}

// ---------------------------------------------------------------------------
extern "C" void kernel_launch(void* const* d_in, const int* in_sizes, int n_in,
                              void* d_out, int out_size, void* d_ws, size_t ws_size,
                              hipStream_t stream) {
  const float* input     = (const float*)d_in[0];  // [T,B,I]
  const float* input_mat = (const float*)d_in[1];  // [I,H]
  const float* blocks    = (const float*)d_in[2];  // [K,bs,bs]
  const float* coupling  = (const float*)d_in[3];  // [H,H]
  const float* out_mat   = (const float*)d_in[4];  // [H,O]

  float* out = (float*)d_out;                                  // [T,B,O]
  float* emb = out + (size_t)T_STEPS * BATCH * OUT_DIM;        // [T,B,H]

  // Workspace layout (well under any plausible ws_size: ~540 KB total):
  //   [0, 16KB)        : nz flags (1024 ints), cnt (32), lst (1024)
  //   [16KB, 16KB+512KB): packed diag weight fragments
  int* nz  = (int*)d_ws;
  int* cnt = nz + KBLK * KBLK;
  int* lst = cnt + KBLK;
  float* pkd = (float*)((char*)d_ws + 16384);

  // 1) inp_proj = input @ input_mat -> written into the embeddings slice.
  {
    int M = T_STEPS * BATCH, K = IN_DIM, N = HDIM;
    int tiles = (M / 32) * (N / 32);
    gemm_wmma_f32<<<tiles / 8, 256, 0, stream>>>(input, input_mat, emb, M, K, N);
  }
  // 2) detect nonzero coupling blocks, build per-column lists, pack weights.
  find_nz_blocks<<<KBLK * KBLK, 256, 0, stream>>>(coupling, nz);
  build_lists<<<1, 32, 0, stream>>>(nz, cnt, lst);
  pack_diag<<<KBLK, 256, 0, stream>>>(blocks, pkd);
  // 3) sequential scan (single WGP-resident workgroup, state in LDS).
  scan_kernel<<<1, 1024, 0, stream>>>(pkd, coupling, emb, cnt, lst);
  // 4) output = embeddings @ out_mat.
  {
    int M = T_STEPS * BATCH, K = HDIM, N = OUT_DIM;
    int tiles = (M / 32) * (N / 32);
    gemm_wmma_f32<<<tiles / 8, 256, 0, stream>>>(emb, out_mat, out, M, K, N);
  }
}